// ClockworkGatedRNN_13795434954914
// MI455X (gfx1250) — compile-verified
//
#include <hip/hip_runtime.h>
#include <hip/hip_bf16.h>
#include <math.h>

typedef __attribute__((ext_vector_type(2))) float v2f;
typedef __attribute__((ext_vector_type(8))) float v8f;

#define B_DIM 64
#define T_DIM 2048
#define DIN   256
#define DOUT  512
#define NSUB  128   // per-group width (DOUT / 4 periods)

__device__ __forceinline__ v8f wmma_f32x4(v2f a, v2f b, v8f c) {
  // V_WMMA_F32_16X16X4_F32 : D = A(16x4,f32) x B(4x16,f32) + C(16x16,f32)
  return __builtin_amdgcn_wmma_f32_16x16x4_f32(
      /*neg_a=*/false, a, /*neg_b=*/false, b,
      /*c_mod=*/(short)0, c, /*reuse_a=*/false, /*reuse_b=*/false);
}

__global__ __launch_bounds__(256) void cw_rnn_final_kernel(
    const float* __restrict__ X,     // (64, 2048, 256)
    const float* __restrict__ W,     // (256, 512)
    const float* __restrict__ bias,  // (512)
    const float* __restrict__ U,     // (4, 128, 128) clock_u
    const float* __restrict__ G,     // (4, 128, 128) clock_gates
    float* __restrict__ out)         // (64, 512)
{
  __shared__ float xsub[B_DIM * NSUB];  // 32 KB: x_sub for this group

  const int g    = blockIdx.x;                    // clock group, period = 1<<g
  const int p    = 1 << g;
  const int tg   = (T_DIM - 1) - ((T_DIM - 1) % p);  // last updating timestep
  const int gcol = g * NSUB;

  const int lane  = threadIdx.x & 31;
  const int wv    = threadIdx.x >> 5;   // 8 waves
  const int n15   = lane & 15;          // M index (A) / N index (B,C,D)
  const int koff  = (lane >> 4) << 1;   // K sub-offset: 0 or 2
  const int mhalf = (lane >> 4) << 3;   // C/D row half: 0 or 8

  // ---------- Stage 1: xsub = X[:, tg, :] @ W[:, gcol:gcol+128] + b ----------
  // 64x128 output -> 4x8 = 32 tiles of 16x16, 4 tiles per wave, K=256.
  for (int tile = wv; tile < 32; tile += 8) {
    const int mt  = tile >> 3;
    const int nt  = tile & 7;
    const int col = gcol + nt * 16 + n15;
    const float* Xrow = X + ((size_t)(mt * 16 + n15) * T_DIM + tg) * DIN;

    v8f acc = {};
#pragma unroll 8
    for (int k0 = 0; k0 < DIN; k0 += 4) {
      v2f a = *reinterpret_cast<const v2f*>(Xrow + k0 + koff);   // 8B global load
      v2f b;
      b.x = W[(size_t)(k0 + koff)     * DOUT + col];
      b.y = W[(size_t)(k0 + koff + 1) * DOUT + col];
      acc = wmma_f32x4(a, b, acc);
    }
    const float bv = bias[col];
#pragma unroll
    for (int v = 0; v < 8; ++v)
      xsub[(mt * 16 + mhalf + v) * NSUB + nt * 16 + n15] = acc[v] + bv;
  }
  __syncthreads();

  // ---------- Stage 2: k = xsub@U_g ; q = (xsub*xsub)@G_g ; elementwise ----------
  const float* Ug = U + (size_t)g * NSUB * NSUB;
  const float* Gg = G + (size_t)g * NSUB * NSUB;

  for (int tile = wv; tile < 32; tile += 8) {
    const int mt   = tile >> 3;
    const int nt   = tile & 7;
    const int colg = nt * 16 + n15;               // column within group
    const float* arow = xsub + (mt * 16 + n15) * NSUB;

    v8f kacc = {};
    v8f qacc = {};
#pragma unroll 8
    for (int k0 = 0; k0 < NSUB; k0 += 4) {
      v2f a   = *reinterpret_cast<const v2f*>(arow + k0 + koff);  // LDS 8B read
      v2f asq = a * a;
      v2f bu, bg;
      bu.x = Ug[(size_t)(k0 + koff)     * NSUB + colg];
      bu.y = Ug[(size_t)(k0 + koff + 1) * NSUB + colg];
      bg.x = Gg[(size_t)(k0 + koff)     * NSUB + colg];
      bg.y = Gg[(size_t)(k0 + koff + 1) * NSUB + colg];
      kacc = wmma_f32x4(a,   bu, kacc);
      qacc = wmma_f32x4(asq, bg, qacc);
    }

#pragma unroll
    for (int v = 0; v < 8; ++v) {
      const int   row = mt * 16 + mhalf + v;       // batch index
      const float xs  = xsub[row * NSUB + colg];
      // z = hard_sigmoid(x_sub + k)
      float z = 0.2f * (xs + kacc[v]) + 0.5f;
      z = fminf(fmaxf(z, 0.0f), 1.0f);
      // z_gate = tanh(q) ; z_out = softplus(x_sub * z_gate)
      const float zg  = tanhf(qacc[v]);
      const float t   = xs * zg;
      const float zo  = (t > 0.0f) ? (t + log1pf(expf(-t))) : log1pf(expf(t));
      out[(size_t)row * DOUT + gcol + colg] = z * zo + (1.0f - z) * xs;
    }
  }
}

extern "C" void kernel_launch(void* const* d_in, const int* in_sizes, int n_in,
                              void* d_out, int out_size, void* d_ws, size_t ws_size,
                              hipStream_t stream) {
  // setup_inputs order: X, W, b, W_gate, b_gate, clock_u, clock_gates
  const float* X    = (const float*)d_in[0];
  const float* W    = (const float*)d_in[1];
  const float* bias = (const float*)d_in[2];
  // d_in[3] (W_gate) and d_in[4] (b_gate) are dead in the reference.
  const float* U    = (const float*)d_in[5];
  const float* G    = (const float*)d_in[6];
  float* out = (float*)d_out;
  (void)in_sizes; (void)n_in; (void)out_size; (void)d_ws; (void)ws_size;

  // One block per clock group; 8 wave32 waves per block.
  cw_rnn_final_kernel<<<dim3(4), dim3(256), 0, stream>>>(X, W, bias, U, G, out);
}